// HashGridPlanesEncoder_21809843929779
// MI455X (gfx1250) — compile-verified
//
#include <hip/hip_runtime.h>
#include <cmath>

// ---------------- HashGridPlanesEncoder (hybrid k-planes), MI455X ----------------
// 1M pts x 3 planes x 16 levels x 4 corners bilinear hash-grid gather.
// Gather-bound: 66MB of tables fit in the 192MB L2 (RT-hint gathers keep them
// resident); 128MB output + 12MB points are streamed with NT hints.
// Hash/dense split per level is static: hash iff r >= 724; resolutions are
// {16,23,31,43,59,81,112,154,213,295,407,562 | 777,1073,1483,2048}, so levels
// 0-11 dense, 12-15 hash (margins ~160/~50 around the threshold -> safe to bake).
// Hash-level tables are always exactly 2^19 entries -> mask 0x7FFFF.

#define NLVL 16
#define FIRST_HASH_LEVEL 12
#define PRIME1 2654435761u
#define HMASK 0x7FFFFu

struct EncParams {
  float    scale[NLVL];     // f32 cast of double scale, as reference
  unsigned stride1[NLVL];   // res + 1 (dense linear stride)
  unsigned offset[NLVL];    // cumulative table offsets (entries)
};

typedef float v2f __attribute__((ext_vector_type(2)));
typedef float v4f __attribute__((ext_vector_type(4)));

__device__ __forceinline__ void plane_accum(float ux, float uy,
                                            const v2f* __restrict__ tab,
                                            const EncParams& P,
                                            v2f acc[NLVL]) {
#pragma unroll
  for (int l = 0; l < NLVL; ++l) {
    const float s  = P.scale[l];
    const float px = fmaf(ux, s, 0.5f);        // pos = u*scale + 0.5
    const float py = fmaf(uy, s, 0.5f);
    const float gx = floorf(px);
    const float gy = floorf(py);
    const float fx = px - gx;                   // frac
    const float fy = py - gy;
    const unsigned x0 = (unsigned)gx;
    const unsigned y0 = (unsigned)gy;
    const unsigned x1 = x0 + 1u;
    const unsigned y1 = y0 + 1u;

    unsigned i00, i10, i01, i11;
    if (l >= FIRST_HASH_LEVEL) {                // compile-time: spatial hash path
      const unsigned h0 = y0 * PRIME1;
      const unsigned h1 = y1 * PRIME1;
      i00 = (x0 ^ h0) & HMASK;  i10 = (x1 ^ h0) & HMASK;
      i01 = (x0 ^ h1) & HMASK;  i11 = (x1 ^ h1) & HMASK;
    } else {                                    // compile-time: dense linear path
      const unsigned st = P.stride1[l];
      const unsigned b0 = y0 * st;
      const unsigned b1 = y1 * st;
      i00 = x0 + b0;  i10 = x1 + b0;
      i01 = x0 + b1;  i11 = x1 + b1;
    }
    const unsigned off = P.offset[l];
    // 4 independent b64 gathers; default RT hint keeps tables in L2.
    const v2f f00 = tab[off + i00];
    const v2f f10 = tab[off + i10];
    const v2f f01 = tab[off + i01];
    const v2f f11 = tab[off + i11];

    const float w00 = (1.f - fx) * (1.f - fy);
    const float w10 = fx * (1.f - fy);
    const float w01 = (1.f - fx) * fy;
    const float w11 = fx * fy;
    // Packed f32 math -> v_pk_fma_f32 (4 packed FMAs instead of 8 scalar).
    acc[l] += w00 * f00 + w10 * f10 + w01 * f01 + w11 * f11;
  }
}

__global__ __launch_bounds__(256)
void hashgrid_kplanes_kernel(const float* __restrict__ pts,
                             const v2f* __restrict__ txy,
                             const v2f* __restrict__ txz,
                             const v2f* __restrict__ tyz,
                             float* __restrict__ out,
                             EncParams P, int n) {
  const int i = blockIdx.x * blockDim.x + threadIdx.x;
  if (i >= n) return;

  // Prefetch point data a few blocks ahead (gfx1250 global_prefetch_b8).
  const int pf = i + 256 * 64;
  if (pf < n) __builtin_prefetch(pts + 3 * (size_t)pf, 0, 0);

  // Streamed-once point data: non-temporal loads.
  const float x = __builtin_nontemporal_load(pts + 3 * (size_t)i + 0);
  const float y = __builtin_nontemporal_load(pts + 3 * (size_t)i + 1);
  const float z = __builtin_nontemporal_load(pts + 3 * (size_t)i + 2);
  const float ux = (x + 1.0f) * 0.5f;
  const float uy = (y + 1.0f) * 0.5f;
  const float uz = (z + 1.0f) * 0.5f;

  v2f acc[NLVL];
#pragma unroll
  for (int k = 0; k < NLVL; ++k) acc[k] = (v2f){0.0f, 0.0f};

  plane_accum(ux, uy, txy, P, acc);   // plane (0,1)
  plane_accum(ux, uz, txz, P, acc);   // plane (0,2)
  plane_accum(uy, uz, tyz, P, acc);   // plane (1,2)

  // 128B/point output, never re-read: NT b128 stores so the 128MB stream
  // does not evict the L2-resident tables.
  v4f* o = (v4f*)(out + (size_t)i * (2 * NLVL));
#pragma unroll
  for (int k = 0; k < 8; ++k) {
    v4f v = { acc[2 * k][0], acc[2 * k][1], acc[2 * k + 1][0], acc[2 * k + 1][1] };
    __builtin_nontemporal_store(v, o + k);
  }
}

// Host-side mirror of the reference's _level_constants(), in double precision.
static void compute_params(EncParams* P) {
  const double pls = pow(2.0, log2(2048.0 / 16.0) / 15.0);
  unsigned long long off = 0;
  for (int l = 0; l < NLVL; ++l) {
    const double s = 16.0 * pow(pls, (double)l) - 1.0;
    const long long r = (long long)ceil(s) + 1;
    const long long dense = (r + 1) * (r + 1);
    long long p = dense < (1LL << 19) ? dense : (1LL << 19);
    p = ((p + 7) / 8) * 8;                       // round up to multiple of 8
    P->scale[l]   = (float)s;
    P->stride1[l] = (unsigned)(r + 1);
    P->offset[l]  = (unsigned)off;
    off += (unsigned long long)p;
  }
}

extern "C" void kernel_launch(void* const* d_in, const int* in_sizes, int n_in,
                              void* d_out, int out_size, void* d_ws, size_t ws_size,
                              hipStream_t stream) {
  (void)n_in; (void)d_ws; (void)ws_size; (void)out_size;
  const float* pts = (const float*)d_in[0];
  const v2f*   txy = (const v2f*)d_in[1];
  const v2f*   txz = (const v2f*)d_in[2];
  const v2f*   tyz = (const v2f*)d_in[3];
  float* out = (float*)d_out;

  EncParams P;
  compute_params(&P);

  const int n = in_sizes[0] / 3;                 // N_PTS
  const int block = 256;                         // 8 wave32 waves
  const int grid = (n + block - 1) / block;
  hashgrid_kplanes_kernel<<<grid, block, 0, stream>>>(pts, txy, txz, tyz, out, P, n);
}